// PointSetAnchorPoseHead_23364622090730
// MI455X (gfx1250) — compile-verified
//
#include <hip/hip_runtime.h>
#include <hip/hip_bf16.h>
#include <math.h>

// ---------------------------------------------------------------------------
// PointSetAnchorPoseHead on MI455X (gfx1250, wave32).
//   K1: 5x5 NMS (LDS-tiled, halo 2)           -> heat_sup (workspace)
//   K2: per-joint, 32 blocks x 8192 elems: iterative top-30 extraction in LDS
//   K3: per-joint merge 960 -> top-30, gather offsets -> candidates (padded 32)
//   K4: nearest-candidate via V_WMMA_F32_16X16X4_F32:
//         argmin_c |p-c|^2 == argmin_c (|c|^2 - 2 p.c); p.c computed as a
//         (16 poses x K=2) x (K=2 x 16 candidates) f32 WMMA; argmin via
//         fully branchless shfl_xor butterfly inside each 16-lane half
//         (wave32 D layout: VGPR v -> rows v / v+8, lane -> column N).
// Memory-bound problem (~45 MB total @ 23.3 TB/s => ~2us floor); all stages
// stream each byte once; candidates live in LDS; all WMMA operand fetches and
// reductions are branchless (cndmask, no exec-mask branches).
// ---------------------------------------------------------------------------

typedef __attribute__((ext_vector_type(2))) float v2f;
typedef __attribute__((ext_vector_type(8))) float v8f;

#define NJ 17
#define HH 512
#define WW 512
#define HW (HH * WW)
#define TOPK 30
#define NCAND 32          // padded to 2 WMMA column tiles
#define PART_BLOCKS 32
#define PART_ELEMS 8192   // 262144 / 32
#define PART_CANDS (PART_BLOCKS * TOPK)  // 960

// ------------------------------ K1: NMS ------------------------------------
__global__ void nms5x5_kernel(const float* __restrict__ heat,
                              float* __restrict__ heat_sup) {
  __shared__ float t[20][20];
  const int j = blockIdx.z;
  const int ox = blockIdx.x * 16, oy = blockIdx.y * 16;
  const int tid = threadIdx.x;
  const int tx = tid & 15, ty = tid >> 4;
  const float* hj = heat + (size_t)j * HW;
  for (int i = tid; i < 400; i += 256) {
    int ly = i / 20, lx = i - ly * 20;
    int gx = ox + lx - 2, gy = oy + ly - 2;
    float v = -INFINITY;
    if (gx >= 0 && gx < WW && gy >= 0 && gy < HH) v = hj[gy * WW + gx];
    t[ly][lx] = v;
  }
  __syncthreads();
  const float c = t[ty + 2][tx + 2];
  float m = -INFINITY;
#pragma unroll
  for (int dy = 0; dy < 5; ++dy)
#pragma unroll
    for (int dx = 0; dx < 5; ++dx)
      m = fmaxf(m, t[ty + dy][tx + dx]);
  heat_sup[(size_t)j * HW + (size_t)(oy + ty) * WW + (ox + tx)] =
      (m == c) ? c : 0.0f;
}

// --------------------- K2: per-block top-30 extraction ---------------------
__global__ void topk_partial_kernel(const float* __restrict__ heat_sup,
                                    float* __restrict__ pvals,
                                    int* __restrict__ pidx) {
  __shared__ float lv[PART_ELEMS];
  __shared__ float rv[256];
  __shared__ int ri[256];
  const int j = blockIdx.y, blk = blockIdx.x, tid = threadIdx.x;
  const int base = blk * PART_ELEMS;
  const float* src = heat_sup + (size_t)j * HW + base;
  for (int i = tid; i < PART_ELEMS; i += 256) lv[i] = src[i];
  __syncthreads();
  const int chunk = tid * 32;
  for (int r = 0; r < TOPK; ++r) {
    float best = lv[chunk];
    int bi = chunk;
#pragma unroll 8
    for (int s = 1; s < 32; ++s) {
      float v = lv[chunk + s];
      if (v > best) { best = v; bi = chunk + s; }  // ascending scan => lowest idx on tie
    }
    rv[tid] = best; ri[tid] = bi;
    __syncthreads();
    for (int s = 128; s > 0; s >>= 1) {
      if (tid < s) {
        float ov = rv[tid + s]; int oi = ri[tid + s];
        if (ov > rv[tid] || (ov == rv[tid] && oi < ri[tid])) { rv[tid] = ov; ri[tid] = oi; }
      }
      __syncthreads();
    }
    if (tid == 0) {
      int sel = ri[0];
      pvals[(j * PART_BLOCKS + blk) * TOPK + r] = rv[0];
      pidx[(j * PART_BLOCKS + blk) * TOPK + r] = base + sel;
      lv[sel] = -INFINITY;
    }
    __syncthreads();
  }
}

// ------------- K3: merge 960 -> top-30, gather offsets, pad to 32 ----------
__global__ void topk_final_kernel(const float* __restrict__ pvals,
                                  const int* __restrict__ pidx,
                                  const float* __restrict__ offset_pred,
                                  const int* __restrict__ stride_p,
                                  float* __restrict__ cand_x,
                                  float* __restrict__ cand_y,
                                  float* __restrict__ cand_n) {
  __shared__ float cv[PART_CANDS];
  __shared__ int ci[PART_CANDS];
  __shared__ float rv[256];
  __shared__ int ri[256];
  const int j = blockIdx.x, tid = threadIdx.x;
  const float fs = (float)stride_p[0];
  for (int i = tid; i < PART_CANDS; i += 256) {
    cv[i] = pvals[j * PART_CANDS + i];
    ci[i] = pidx[j * PART_CANDS + i];
  }
  __syncthreads();
  const float* offj = offset_pred + (size_t)j * 2 * HW;
  for (int r = 0; r < TOPK; ++r) {
    float best = cv[tid];
    int bslot = tid;
    for (int i = tid + 256; i < PART_CANDS; i += 256) {
      float v = cv[i];
      if (v > best || (v == best && ci[i] < ci[bslot])) { best = v; bslot = i; }
    }
    rv[tid] = best; ri[tid] = bslot;
    __syncthreads();
    for (int s = 128; s > 0; s >>= 1) {
      if (tid < s) {
        float ov = rv[tid + s]; int os = ri[tid + s];
        if (ov > rv[tid] || (ov == rv[tid] && ci[os] < ci[ri[tid]])) { rv[tid] = ov; ri[tid] = os; }
      }
      __syncthreads();
    }
    if (tid == 0) {
      int slot = ri[0];
      int q = ci[slot];
      float x = (float)(q & (WW - 1));
      float y = (float)(q >> 9);
      float hx = fs * (x + offj[q]);
      float hy = fs * (y + offj[HW + q]);
      cand_x[j * NCAND + r] = hx;
      cand_y[j * NCAND + r] = hy;
      cand_n[j * NCAND + r] = hx * hx + hy * hy;
      cv[slot] = -INFINITY;
    }
    __syncthreads();
  }
  if (tid >= TOPK && tid < NCAND) {  // padding columns: never win argmin
    cand_x[j * NCAND + tid] = 0.0f;
    cand_y[j * NCAND + tid] = 0.0f;
    cand_n[j * NCAND + tid] = 3.0e38f;
  }
}

// --------------- K4: nearest candidate per pose/joint via WMMA -------------
__global__ void refine_wmma_kernel(const float* __restrict__ poses,
                                   const float* __restrict__ cand_x,
                                   const float* __restrict__ cand_y,
                                   const float* __restrict__ cand_n,
                                   float* __restrict__ out, int N) {
  __shared__ float scx[NJ * NCAND];
  __shared__ float scy[NJ * NCAND];
  __shared__ float scn[NJ * NCAND];
  const int tid = threadIdx.x;
  for (int i = tid; i < NJ * NCAND; i += 256) {
    scx[i] = cand_x[i];
    scy[i] = cand_y[i];
    scn[i] = cand_n[i];
  }
  __syncthreads();
  const int wave = tid >> 5, lane = tid & 31;
  const int tile = blockIdx.x * 8 + wave;
  const int tiles = (N + 15) >> 4;
  if (tile >= tiles) return;  // wave-uniform: EXEC stays all-1s for WMMA
  const int base = tile * 16;
  const bool lo = lane < 16;
  const int n16 = lane & 15;
  // Branchless A row fetch: every lane loads a valid (clamped) pose row;
  // lanes 16-31 and OOB rows are zeroed with cndmask (no exec branches).
  const int prow = min(base + n16, N - 1);
  const bool avalid = lo && (base + n16 < N);

  for (int j = 0; j < NJ; ++j) {
    // A (16x4 f32): lanes 0-15 hold K=0 (px) / K=1 (py); lanes 16-31 -> K=2,3 = 0
    const float ax = poses[(size_t)prow * (2 * NJ) + 2 * j];
    const float ay = poses[(size_t)prow * (2 * NJ) + 2 * j + 1];
    v2f a;
    a[0] = avalid ? ax : 0.0f;
    a[1] = avalid ? ay : 0.0f;
    float bestv[8];
    int besti[8];
#pragma unroll
    for (int g = 0; g < 2; ++g) {
      // B (4x16 f32): row K=0 = cx over N (lanes 0-15), K=1 = cy; K=2,3 = 0.
      // LDS address (lane&15) is valid for all lanes -> unconditional load,
      // hi half zeroed by select.
      const float bx = scx[j * NCAND + n16 + g * 16];
      const float by = scy[j * NCAND + n16 + g * 16];
      const float cn = scn[j * NCAND + n16 + g * 16];
      v2f b;
      b[0] = lo ? bx : 0.0f;
      b[1] = lo ? by : 0.0f;
      v8f c = {0.f, 0.f, 0.f, 0.f, 0.f, 0.f, 0.f, 0.f};
      v8f d = __builtin_amdgcn_wmma_f32_16x16x4_f32(
          false, a, false, b, (short)0, c, false, false);
#pragma unroll
      for (int v = 0; v < 8; ++v) {
        float s = cn - 2.0f * d[v];  // |c|^2 - 2 p.c  (|p|^2 constant per row)
        if (g == 0) {
          bestv[v] = s;
          besti[v] = n16;
        } else {
          const bool take = s < bestv[v];  // group 0 wins ties (lower index)
          bestv[v] = take ? s : bestv[v];
          besti[v] = take ? (n16 + 16) : besti[v];
        }
      }
    }
    // D layout: VGPR v -> row v (lanes 0-15) / row v+8 (lanes 16-31), N = lane&15.
    // Branchless butterfly argmin across each 16-lane half; lowest index wins
    // ties (non-short-circuit ops keep this pure v_cmp/v_cndmask, no branches).
#pragma unroll
    for (int v = 0; v < 8; ++v) {
      float val = bestv[v];
      int idx = besti[v];
#pragma unroll
      for (int m = 1; m <= 8; m <<= 1) {
        const float ov = __shfl_xor(val, m, 32);
        const int oi = __shfl_xor(idx, m, 32);
        const bool lt = ov < val;
        const bool eq = ov == val;
        const bool ilt = oi < idx;
        const bool take = lt | (eq & ilt);
        val = take ? ov : val;
        idx = take ? oi : idx;
      }
      if (n16 == 0) {
        int row = v + ((lane >> 4) << 3);
        int p = base + row;
        if (p < N) {
          out[(size_t)p * (2 * NJ) + 2 * j] = scx[j * NCAND + idx];
          out[(size_t)p * (2 * NJ) + 2 * j + 1] = scy[j * NCAND + idx];
        }
      }
    }
  }
}

// ------------------------------ launcher -----------------------------------
extern "C" void kernel_launch(void* const* d_in, const int* in_sizes, int n_in,
                              void* d_out, int out_size, void* d_ws, size_t ws_size,
                              hipStream_t stream) {
  const float* poses = (const float*)d_in[0];
  const float* heat = (const float*)d_in[1];
  const float* offset = (const float*)d_in[2];
  const int* stride_p = (const int*)d_in[3];
  const int N = in_sizes[0] / (2 * NJ);

  char* ws = (char*)d_ws;
  size_t off = 0;
  float* heat_sup = (float*)(ws + off); off += (size_t)NJ * HW * sizeof(float);
  float* pvals = (float*)(ws + off);    off += (size_t)NJ * PART_CANDS * sizeof(float);
  int* pidx = (int*)(ws + off);         off += (size_t)NJ * PART_CANDS * sizeof(int);
  float* cand_x = (float*)(ws + off);   off += (size_t)NJ * NCAND * sizeof(float);
  float* cand_y = (float*)(ws + off);   off += (size_t)NJ * NCAND * sizeof(float);
  float* cand_n = (float*)(ws + off);   off += (size_t)NJ * NCAND * sizeof(float);

  nms5x5_kernel<<<dim3(WW / 16, HH / 16, NJ), 256, 0, stream>>>(heat, heat_sup);
  topk_partial_kernel<<<dim3(PART_BLOCKS, NJ), 256, 0, stream>>>(heat_sup, pvals, pidx);
  topk_final_kernel<<<NJ, 256, 0, stream>>>(pvals, pidx, offset, stride_p,
                                            cand_x, cand_y, cand_n);
  const int tiles = (N + 15) / 16;
  const int blocks = (tiles + 7) / 8;
  refine_wmma_kernel<<<blocks, 256, 0, stream>>>(poses, cand_x, cand_y, cand_n,
                                                 (float*)d_out, N);
}